// CentroidUOMNet_70403103916540
// MI455X (gfx1250) — compile-verified
//
#include <hip/hip_runtime.h>
#include <math.h>

// MI455X / gfx1250, wave32. Full-precision matrix path: V_WMMA_F32_16X16X4_F32.
typedef __attribute__((ext_vector_type(2))) float v2f;
typedef __attribute__((ext_vector_type(8))) float v8f;

#define N_EDGE 4096
#define DEG    32
#define KS     8
#define D0     128
#define NC     40
#define WAVES  4
#define THREADS (WAVES * 32)

// workspace layout (floats): weights pre-swizzled into WMMA fragment order
#define WRE_C0 0         // [3][32][2][128][2]  conv1 weights, 49152 floats
#define WRE_C1 49152     // same for conv2
#define WRE_U0 98304     // [32][2][64][2]      uom1 weights, 8192 floats
#define WRE_U1 106496    // same for uom2
#define WS_FLOATS 114688

static __device__ __forceinline__ v8f wmma4(v2f a, v2f b, v8f c) {
  // D = A(16x4 f32) * B(4x16 f32) + C(16x16 f32)
  return __builtin_amdgcn_wmma_f32_16x16x4_f32(
      /*neg_a=*/false, a, /*neg_b=*/false, b,
      /*c_mod=*/(short)0, c, /*reuse_a=*/false, /*reuse_b=*/false);
}

static __device__ __forceinline__ v8f zero8() {
  v8f z = {0.f, 0.f, 0.f, 0.f, 0.f, 0.f, 0.f, 0.f};
  return z;
}

// Wave-local LDS producer->consumer handoff (cross-lane, same wave):
// CDNA5 split counters -- wait for all outstanding DS ops of this wave.
static __device__ __forceinline__ void wsync() {
  __builtin_amdgcn_wave_barrier();
  asm volatile("s_wait_dscnt 0" ::: "memory");
  __builtin_amdgcn_wave_barrier();
}

// ---- unorder-mapping GEMM: T = relu(SE @ Wu + bu), mbuf = mean_k(T) ----
// se  : LDS, two stacked blocks, flat [2][8][128]; A row m -> se[m*128 + f]
// mbuf: LDS [2][64]   (mbuf[j][i*8+jj] = m_j[i][jj])
template <bool PRE>
static __device__ __forceinline__ void uom_mstep(const float* __restrict__ se,
                                                 const float* __restrict__ Wu,
                                                 const float* __restrict__ WuT,
                                                 const float* __restrict__ bu,
                                                 float* __restrict__ mbuf,
                                                 int lane) {
  const int lr  = lane & 15;
  const int hi  = lane >> 4;
  const int kk0 = hi ? 2 : 0;
  v8f acc[4];
  acc[0] = zero8(); acc[1] = zero8(); acc[2] = zero8(); acc[3] = zero8();
  for (int ks = 0; ks < 32; ++ks) {     // K = 128 features, 4 per step
    const int f0 = ks * 4 + kk0;
    v2f a;
    a.x = se[lr * 128 + f0];            // A[m=lr][f0]   ((j*1024+k*128)==lr*128)
    a.y = se[lr * 128 + f0 + 1];
#pragma unroll
    for (int nt = 0; nt < 4; ++nt) {    // 64 output cols = 4 N-tiles
      const int c = nt * 16 + lr;
      v2f b;
      if (PRE) {                        // fragment-order: one coalesced b64/lane
        b = ((const v2f*)WuT)[(ks * 2 + hi) * 64 + c];
      } else {
        b.x = Wu[f0 * 64 + c];
        b.y = Wu[(f0 + 1) * 64 + c];
      }
      acc[nt] = wmma4(a, b, acc[nt]);
    }
  }
  // rows 0-7 (block0) in lanes 0-15, rows 8-15 (block1) in lanes 16-31:
  // mean over the 8 sample-rows is a per-lane reduction over the 8 acc elems.
#pragma unroll
  for (int nt = 0; nt < 4; ++nt) {
    const int c = nt * 16 + lr;
    const float bias = bu[c];
    float s = 0.f;
#pragma unroll
    for (int v = 0; v < 8; ++v) s += fmaxf(acc[nt][v] + bias, 0.f);
    mbuf[hi * 64 + c] = s * 0.125f;
  }
}

// ---- UE = blockdiag(m0,m1) @ SE_stacked ; overwrites se with ue ----
static __device__ __forceinline__ void uom_uestep(const float* __restrict__ mbuf,
                                                  float* __restrict__ se,
                                                  int lane) {
  const int lr  = lane & 15;
  const int hi  = lane >> 4;
  const int kk0 = hi ? 2 : 0;
  v8f acc[8];
#pragma unroll
  for (int nt = 0; nt < 8; ++nt) acc[nt] = zero8();
  for (int ks2 = 0; ks2 < 4; ++ks2) {   // K = 16 (stacked), 4 per step
    const int k0 = ks2 * 4 + kk0;
    const int k1 = k0 + 1;
    v2f a;  // block-diagonal mixing matrix
    a.x = ((lr >> 3) == (k0 >> 3)) ? mbuf[(lr >> 3) * 64 + (lr & 7) * 8 + (k0 & 7)] : 0.f;
    a.y = ((lr >> 3) == (k1 >> 3)) ? mbuf[(lr >> 3) * 64 + (lr & 7) * 8 + (k1 & 7)] : 0.f;
#pragma unroll
    for (int nt = 0; nt < 8; ++nt) {    // 128 feature cols = 8 N-tiles
      v2f b;
      b.x = se[k0 * 128 + nt * 16 + lr];
      b.y = se[k1 * 128 + nt * 16 + lr];
      acc[nt] = wmma4(a, b, acc[nt]);
    }
  }
  wsync();                              // all SE reads done before overwrite
#pragma unroll
  for (int nt = 0; nt < 8; ++nt)
#pragma unroll
    for (int v = 0; v < 8; ++v) {
      const int row = v + hi * 8;       // D row -> (block=row>>3, k=row&7)
      se[row * 128 + nt * 16 + lr] = acc[nt][v];
    }
}

// ---- conv1d (3-tap VALID) as GEMM with tap-shifted B columns ----
// ue (LDS, [2][8][128]) reinterpreted per block as agg[128][8] (raw reshape).
// C cols: n = block*8 + l ; cols with l>=6 are padding (zeros).
template <bool PRE>
static __device__ __forceinline__ v8f conv_tile(const float* __restrict__ ue,
                                                const float* __restrict__ Wc,
                                                const float* __restrict__ Wre,
                                                int mt, int lane) {
  const int lr  = lane & 15;
  const int hi  = lane >> 4;
  const int kk0 = hi ? 2 : 0;
  const int l   = lr & 7;
  const bool valid = (l < 6);
  const float* ueflat = ue + (lr >> 3) * 1024;  // this block's agg, flat
  const int o = mt * 16 + lr;
  v8f acc = zero8();
  for (int t = 0; t < 3; ++t) {
    const int p = l + t;                // shifted column (<= 7)
    for (int ks = 0; ks < 32; ++ks) {   // K = 128 in-channels, 4 per step
      const int i0 = ks * 4 + kk0;
      v2f a, b;
      if (PRE) {                        // fragment-order: one coalesced b64/lane
        a = ((const v2f*)Wre)[((t * 32 + ks) * 2 + hi) * 128 + o];
      } else {                          // Wc layout [128][128][3]
        a.x = Wc[o * 384 + i0 * 3 + t];
        a.y = Wc[o * 384 + (i0 + 1) * 3 + t];
      }
      b.x = valid ? ueflat[i0 * 8 + p] : 0.f;
      b.y = valid ? ueflat[(i0 + 1) * 8 + p] : 0.f;
      acc = wmma4(a, b, acc);
    }
  }
  return acc;
}

// ---- prep: swizzle weights into WMMA fragment order (coalesced b64 loads) ----
__global__ void prep_weights(const float* __restrict__ Wu0, const float* __restrict__ Wu1,
                             const float* __restrict__ Wc0, const float* __restrict__ Wc1,
                             float* __restrict__ ws) {
  const int i = blockIdx.x * blockDim.x + threadIdx.x;
  if (i < 49152) {  // flat = t*16384 + ks*512 + h*256 + m*2 + e
    const int e  = i & 1;
    const int m  = (i >> 1) & 127;
    const int h  = (i >> 8) & 1;
    const int ks = (i >> 9) & 31;
    const int t  = i >> 14;
    const int src = m * 384 + (ks * 4 + h * 2 + e) * 3 + t;
    ws[WRE_C0 + i] = Wc0[src];
    ws[WRE_C1 + i] = Wc1[src];
  }
  if (i < 8192) {   // flat = ks*256 + h*128 + c*2 + e
    const int e  = i & 1;
    const int c  = (i >> 1) & 63;
    const int h  = (i >> 7) & 1;
    const int ks = i >> 8;
    const int src = (ks * 4 + h * 2 + e) * 64 + c;
    ws[WRE_U0 + i] = Wu0[src];
    ws[WRE_U1 + i] = Wu1[src];
  }
}

template <bool PRE>
__global__ __launch_bounds__(THREADS)
void centroid_uom_fused(const float* __restrict__ feats,
                        const float* __restrict__ Wu0, const float* __restrict__ bu0,
                        const float* __restrict__ Wu1, const float* __restrict__ bu1,
                        const float* __restrict__ Wc0, const float* __restrict__ bc0,
                        const float* __restrict__ Wc1, const float* __restrict__ bc1,
                        const float* __restrict__ Wf,  const float* __restrict__ bf,
                        const int* __restrict__ ids, const int* __restrict__ edge_dict,
                        const int* __restrict__ samp1, const int* __restrict__ samp2,
                        const float* __restrict__ wre,
                        float* __restrict__ out) {
  __shared__ __align__(16) float seue[WAVES][2][KS][D0];  // 32 KB se/ue staging
  __shared__ __align__(16) float scr[WAVES][256];         //  4 KB mbuf / C-tile
  __shared__ __align__(16) float ne[DEG][D0];             // 16 KB node_embs
  __shared__ float embsL[D0];
  __shared__ float redL[NC];

  const int n    = blockIdx.x;
  const int tid  = threadIdx.x;
  const int wv   = tid >> 5;
  const int lane = tid & 31;
  const int id   = ids[n];

  float* seW  = &seue[wv][0][0][0];
  float* scrW = &scr[wv][0];

  // ================= layer 1: 32 member blocks, 2 per wave-task ============
  for (int task = 0; task < DEG / (WAVES * 2); ++task) {
    const int dbase = wv * 8 + task * 2;
    // gather sampled inner-member feature rows (8 x 512B per block, L2-hot)
    for (int j = 0; j < 2; ++j) {
      const int d   = dbase + j;
      const int nbv = edge_dict[id * DEG + d];
      const int* s1 = samp1 + (n * DEG + d) * KS;
      for (int q = lane; q < KS * 32; q += 32) {
        const int k  = q >> 5;
        const int c4 = q & 31;
        const int inner = edge_dict[nbv * DEG + s1[k]];
        const float4 vv = *(const float4*)(feats + (size_t)inner * D0 + c4 * 4);
        *(float4*)(&seue[wv][j][k][c4 * 4]) = vv;
      }
    }
    wsync();
    uom_mstep<PRE>(seW, Wu0, wre + WRE_U0, bu0, scrW, lane);
    wsync();
    uom_uestep(scrW, seW, lane);
    wsync();
    for (int mt = 0; mt < 8; ++mt) {
      v8f y = conv_tile<PRE>(seW, Wc0, wre + WRE_C0, mt, lane);
      const int lr = lane & 15, hi = lane >> 4;
#pragma unroll
      for (int v = 0; v < 8; ++v) scrW[(v + hi * 8) * 16 + lr] = y[v];
      wsync();
      { // per (row, block): relu(+bias) then mean over 6 valid positions
        const int row16 = lane & 15;
        const int j     = lane >> 4;
        const float bias = bc0[mt * 16 + row16];
        float s = 0.f;
#pragma unroll
        for (int l = 0; l < 6; ++l)
          s += fmaxf(scrW[row16 * 16 + j * 8 + l] + bias, 0.f);
        ne[dbase + j][mt * 16 + row16] = s * (1.f / 6.f);
      }
      wsync();
    }
  }
  __syncthreads();

  // ================= layer 2: one block per edge, wave 0 ===================
  if (wv == 0) {
    for (int q = lane; q < KS * D0; q += 32) {   // duplicate into both halves
      const int r = q >> 7, f = q & 127;
      const float val = ne[samp2[n * KS + r]][f];
      seue[0][0][r][f] = val;
      seue[0][1][r][f] = val;
    }
    wsync();
    uom_mstep<PRE>(seW, Wu1, wre + WRE_U1, bu1, scrW, lane);
    wsync();
    uom_uestep(scrW, seW, lane);
    wsync();
    for (int mt = 0; mt < 8; ++mt) {
      v8f y = conv_tile<PRE>(seW, Wc1, wre + WRE_C1, mt, lane);
      const int lr = lane & 15, hi = lane >> 4;
#pragma unroll
      for (int v = 0; v < 8; ++v) scrW[(v + hi * 8) * 16 + lr] = y[v];
      wsync();
      if (lane < 16) {  // log_softmax over length, then mean over length
        const int o = mt * 16 + lane;
        const float bias = bc1[o];
        float z[6], mx = -3.4e38f, sm = 0.f, mn = 0.f;
#pragma unroll
        for (int l = 0; l < 6; ++l) { z[l] = scrW[lane * 16 + l] + bias; mx = fmaxf(mx, z[l]); }
#pragma unroll
        for (int l = 0; l < 6; ++l) { sm += expf(z[l] - mx); mn += z[l]; }
        embsL[o] = mn * (1.f / 6.f) - (mx + logf(sm));
      }
      wsync();
    }
  }
  __syncthreads();

  // ================= classifier + log_softmax over 40 ======================
  if (tid < NC) {
    float acc = bf[tid];
    for (int f = 0; f < D0; ++f) acc += embsL[f] * Wf[f * NC + tid];
    redL[tid] = acc;
  }
  __syncthreads();
  if (tid < NC) {
    float mx = -3.4e38f;
    for (int c = 0; c < NC; ++c) mx = fmaxf(mx, redL[c]);
    float sm = 0.f;
    for (int c = 0; c < NC; ++c) sm += expf(redL[c] - mx);
    out[(size_t)n * NC + tid] = redL[tid] - (mx + logf(sm));
  }
}

extern "C" void kernel_launch(void* const* d_in, const int* in_sizes, int n_in,
                              void* d_out, int out_size, void* d_ws, size_t ws_size,
                              hipStream_t stream) {
  (void)in_sizes; (void)n_in; (void)out_size;
  const float* feats = (const float*)d_in[0];
  const float* Wu0   = (const float*)d_in[1];
  const float* bu0   = (const float*)d_in[2];
  const float* Wu1   = (const float*)d_in[3];
  const float* bu1   = (const float*)d_in[4];
  const float* Wc0   = (const float*)d_in[5];
  const float* bc0   = (const float*)d_in[6];
  const float* Wc1   = (const float*)d_in[7];
  const float* bc1   = (const float*)d_in[8];
  const float* Wf    = (const float*)d_in[9];
  const float* bf    = (const float*)d_in[10];
  const int*   ids   = (const int*)d_in[11];
  const int*   edict = (const int*)d_in[12];
  const int*   samp1 = (const int*)d_in[13];
  const int*   samp2 = (const int*)d_in[14];
  float* out = (float*)d_out;

  const bool pre = (ws_size >= (size_t)WS_FLOATS * sizeof(float)) && d_ws != nullptr;
  if (pre) {
    prep_weights<<<dim3(192), dim3(256), 0, stream>>>(Wu0, Wu1, Wc0, Wc1, (float*)d_ws);
    centroid_uom_fused<true><<<dim3(N_EDGE), dim3(THREADS), 0, stream>>>(
        feats, Wu0, bu0, Wu1, bu1, Wc0, bc0, Wc1, bc1, Wf, bf,
        ids, edict, samp1, samp2, (const float*)d_ws, out);
  } else {
    centroid_uom_fused<false><<<dim3(N_EDGE), dim3(THREADS), 0, stream>>>(
        feats, Wu0, bu0, Wu1, bu1, Wc0, bc0, Wc1, bc1, Wf, bf,
        ids, edict, samp1, samp2, (const float*)d_ws, out);
  }
}